// CRF_35476429865754
// MI455X (gfx1250) — compile-verified
//
#include <hip/hip_runtime.h>
#include <stdint.h>

#define B_SZ 512
#define T_SZ 512
#define K_SZ 128
#define BT   16        // batch rows per workgroup
#define LSTR 132       // LDS row stride (floats), padded for bank spread

typedef float v2f __attribute__((ext_vector_type(2)));
typedef float v8f __attribute__((ext_vector_type(8)));

// ---------------------------------------------------------------------------
// Numerator: gold-path score per batch element (cheap, 1 thread per batch).
// ---------------------------------------------------------------------------
__global__ __launch_bounds__(256) void crf_score_kernel(
    const float* __restrict__ emissions, const int* __restrict__ tags,
    const uint8_t* __restrict__ mask, const float* __restrict__ trans,
    const float* __restrict__ startt, const float* __restrict__ endt,
    float* __restrict__ score_out)
{
    int b = blockIdx.x * blockDim.x + threadIdx.x;
    if (b >= B_SZ) return;
    const int*     tb = tags + (size_t)b * T_SZ;
    const uint8_t* mb = mask + (size_t)b * T_SZ;
    const float*   eb = emissions + (size_t)b * T_SZ * K_SZ;

    float score = startt[tb[0]];
    float msum  = 0.0f;
    int   prev  = tb[0];
    for (int t = 0; t < T_SZ; ++t) {
        float mf = mb[t] ? 1.0f : 0.0f;
        msum += mf;
        int tg = tb[t];
        score += eb[(size_t)t * K_SZ + tg] * mf;
        if (t > 0) score += trans[prev * K_SZ + tg] * mf;
        prev = tg;
    }
    int last_idx = (int)(msum - 1.0f);
    score += endt[tb[last_idx]];
    score_out[b] = score;
}

// ---------------------------------------------------------------------------
// Forward algorithm: exp-space recurrence as f32 WMMA GEMM.
// One WG = 16 batch rows; 8 waves each own 16 output columns and keep their
// exp(trans) B-operand slice (128x16 f32) resident in 64 VGPRs.
// ---------------------------------------------------------------------------
__global__ __launch_bounds__(256) void crf_forward_kernel(
    const float* __restrict__ emissions, const uint8_t* __restrict__ mask,
    const float* __restrict__ trans, const float* __restrict__ startt,
    const float* __restrict__ endt, float* __restrict__ logz_out)
{
    __shared__ float alpha[BT][LSTR];
    __shared__ float P[BT][LSTR];
    __shared__ float partial[BT][16];
    __shared__ float mrow[BT];
    __shared__ float mskrow[BT];

    const int tid  = threadIdx.x;
    const int wave = tid >> 5;
    const int lane = tid & 31;
    const int half = lane >> 4;          // 0: lanes 0-15, 1: lanes 16-31
    const int jcol = wave * 16 + (lane & 15);
    const int b0   = blockIdx.x * BT;

    // --- Preload B operand: exp(trans[:, jcol]) in WMMA 16x16x4 B layout.
    // B is 4x16 per instruction: VGPR pair holds K rows {k0,k0+1} for lanes
    // 0-15 and {k0+2,k0+3} for lanes 16-31, column = jcol.
    v2f breg[32];
    #pragma unroll
    for (int ib = 0; ib < 8; ++ib) {
        #pragma unroll
        for (int ks = 0; ks < 4; ++ks) {
            int k0 = ib * 16 + ks * 4 + half * 2;
            v2f v;
            v.x = __expf(trans[k0 * K_SZ + jcol]);
            v.y = __expf(trans[(k0 + 1) * K_SZ + jcol]);
            breg[ib * 4 + ks] = v;
        }
    }

    // --- alpha0 = start + emissions[:,0,:]
    {
        int row = tid >> 4, sl = tid & 15;
        const float* e = emissions + (size_t)(b0 + row) * T_SZ * K_SZ;
        for (int c = sl; c < K_SZ; c += 16)
            alpha[row][c] = startt[c] + e[c];
    }

    for (int t = 1; t < T_SZ; ++t) {
        __syncthreads();
        // Stage A: partial row maxima (16 lanes per batch row).
        {
            int row = tid >> 4, sl = tid & 15;
            float pm = -3.4e38f;
            for (int c = sl; c < K_SZ; c += 16)
                pm = fmaxf(pm, alpha[row][c]);
            partial[row][sl] = pm;
        }
        __syncthreads();
        // Stage B: finalize row max, stage per-row mask for this t.
        if (tid < BT) {
            float m = partial[tid][0];
            #pragma unroll
            for (int i = 1; i < 16; ++i) m = fmaxf(m, partial[tid][i]);
            mrow[tid]   = m;
            mskrow[tid] = mask[(size_t)(b0 + tid) * T_SZ + t] ? 1.0f : 0.0f;
        }
        __syncthreads();
        // Stage C: P = exp(alpha - m)  (the WMMA A operand source).
        {
            int row = tid >> 4, sl = tid & 15;
            float m = mrow[row];
            for (int c = sl; c < K_SZ; c += 16)
                P[row][c] = __expf(alpha[row][c] - m);
        }
        __syncthreads();

        // --- GEMM: S(16x16 tile) = P(16x128) * expTrans(128x16), K in steps of 4.
        // EXEC is all-ones here (no divergence in this region).
        v8f acc = {0.f, 0.f, 0.f, 0.f, 0.f, 0.f, 0.f, 0.f};
        const int arow = lane & 15;      // A: lanes hold M rows 0-15 in both halves
        #pragma unroll
        for (int ib = 0; ib < 8; ++ib) {
            #pragma unroll
            for (int ks = 0; ks < 4; ++ks) {
                int kb = ib * 16 + ks * 4 + half * 2;
                v2f a;
                a.x = P[arow][kb];
                a.y = P[arow][kb + 1];
                acc = __builtin_amdgcn_wmma_f32_16x16x4_f32(
                    false, a, false, breg[ib * 4 + ks],
                    (short)0, acc, false, false);
            }
        }

        // --- alpha' = mask ? m + log(S) + emit : alpha
        // D layout: VGPR r -> row r (lanes 0-15) / row r+8 (lanes 16-31), col jcol.
        #pragma unroll
        for (int r = 0; r < 8; ++r) {
            int row = r + half * 8;
            const float* ep = emissions +
                (size_t)(b0 + row) * T_SZ * K_SZ + (size_t)t * K_SZ + jcol;
            float ev = ep[0];
            __builtin_prefetch(ep + K_SZ, 0, 0);   // speculative, next timestep
            float nv = mrow[row] + __logf(acc[r]) + ev;
            float ov = alpha[row][jcol];
            float mf = mskrow[row];
            alpha[row][jcol] = mf * nv + (1.0f - mf) * ov;
        }
    }

    // --- log_z = logsumexp(alpha + end_transitions)
    __syncthreads();
    {
        int row = tid >> 4, sl = tid & 15;
        float pm = -3.4e38f;
        for (int c = sl; c < K_SZ; c += 16)
            pm = fmaxf(pm, alpha[row][c] + endt[c]);
        partial[row][sl] = pm;
    }
    __syncthreads();
    if (tid < BT) {
        float m = partial[tid][0];
        #pragma unroll
        for (int i = 1; i < 16; ++i) m = fmaxf(m, partial[tid][i]);
        mrow[tid] = m;
    }
    __syncthreads();
    {
        int row = tid >> 4, sl = tid & 15;
        float m = mrow[row];
        float ps = 0.0f;
        for (int c = sl; c < K_SZ; c += 16)
            ps += __expf(alpha[row][c] + endt[c] - m);
        partial[row][sl] = ps;
    }
    __syncthreads();
    if (tid < BT) {
        float s = 0.0f;
        #pragma unroll
        for (int i = 0; i < 16; ++i) s += partial[tid][i];
        logz_out[b0 + tid] = mrow[tid] + __logf(s);
    }
}

// ---------------------------------------------------------------------------
// Final: nll = -mean(score - log_z)
// ---------------------------------------------------------------------------
__global__ __launch_bounds__(256) void crf_final_kernel(
    const float* __restrict__ score, const float* __restrict__ logz,
    float* __restrict__ out)
{
    __shared__ float red[256];
    int tid = threadIdx.x;
    float s = 0.0f;
    for (int b = tid; b < B_SZ; b += 256)
        s += score[b] - logz[b];
    red[tid] = s;
    __syncthreads();
    for (int off = 128; off > 0; off >>= 1) {
        if (tid < off) red[tid] += red[tid + off];
        __syncthreads();
    }
    if (tid == 0) out[0] = -red[0] / (float)B_SZ;
}

// ---------------------------------------------------------------------------
extern "C" void kernel_launch(void* const* d_in, const int* in_sizes, int n_in,
                              void* d_out, int out_size, void* d_ws, size_t ws_size,
                              hipStream_t stream)
{
    const float*   emissions = (const float*)d_in[0];
    const int*     tags      = (const int*)d_in[1];
    const uint8_t* mask      = (const uint8_t*)d_in[2];
    const float*   trans     = (const float*)d_in[3];
    const float*   startt    = (const float*)d_in[4];
    const float*   endt      = (const float*)d_in[5];

    float* score_ws = (float*)d_ws;          // [B_SZ]
    float* logz_ws  = score_ws + B_SZ;       // [B_SZ]

    crf_score_kernel<<<(B_SZ + 255) / 256, 256, 0, stream>>>(
        emissions, tags, mask, trans, startt, endt, score_ws);
    crf_forward_kernel<<<B_SZ / BT, 256, 0, stream>>>(
        emissions, mask, trans, startt, endt, logz_ws);
    crf_final_kernel<<<1, 256, 0, stream>>>(score_ws, logz_ws, (float*)d_out);
}